// Criterion_13700945674760
// MI455X (gfx1250) — compile-verified
//
#include <hip/hip_runtime.h>

// Problem constants (match the reference: BS=4096, NT=8192).
#define BS_N 4096
#define NT_N 8192

typedef __attribute__((ext_vector_type(2))) float v2f;
typedef __attribute__((ext_vector_type(4))) float v4f;
typedef __attribute__((ext_vector_type(8))) float v8f;

// per-element contribution: log(sigma) + (y-mu)^2 / sigma
// v_log_f32 + v_rcp_f32 (TRANS) + sub + mul + fma + acc-add: ~6 VALU/TRANS ops,
// well under the VALU budget at 23.3 TB/s streaming rate.
__device__ __forceinline__ float elem_f(float m, float s, float y) {
    float d = y - m;
    float r = __builtin_amdgcn_rcpf(s);        // v_rcp_f32, ~1 ulp
    return fmaf(d * d, r, __logf(s));
}

// Wave32 reduction using the CDNA5 matrix pipe.
// A = ones(16x4) (every lane contributes 2 ones -> all 64 entries are 1).
// B(4x16) holds each lane's partial exactly once (VGPR0 = p, VGPR1 = 0),
// so D[m][n] = p_n + p_{n+16} for every m. Lane L then holds q_{L&15} in c[0];
// 4 xor-shuffles over the low 4 index bits finish the 16-way fold.
__device__ __forceinline__ float wave_reduce_wmma(float p) {
    v2f a; a[0] = 1.0f; a[1] = 1.0f;
    v2f b; b[0] = p;    b[1] = 0.0f;
    v8f c = {};
    c = __builtin_amdgcn_wmma_f32_16x16x4_f32(
        /*neg_a=*/false, a, /*neg_b=*/false, b,
        /*c_mod=*/(short)0, c, /*reuse_a=*/false, /*reuse_b=*/false);
    float q = c[0];
    q += __shfl_xor(q, 1, 32);
    q += __shfl_xor(q, 2, 32);
    q += __shfl_xor(q, 4, 32);
    q += __shfl_xor(q, 8, 32);
    return q;   // every lane: full wave sum
}

__global__ void gauss_nll_partial(const float* __restrict__ mu,
                                  const float* __restrict__ sigma,
                                  const float* __restrict__ ty,
                                  float* __restrict__ partials,
                                  int n) {
    const v4f* mu4 = reinterpret_cast<const v4f*>(mu);
    const v4f* sg4 = reinterpret_cast<const v4f*>(sigma);
    const v4f* ty4 = reinterpret_cast<const v4f*>(ty);

    const long long n4     = (long long)(n >> 2);
    const long long stride = (long long)gridDim.x * (long long)blockDim.x;

    float p = 0.0f;
    for (long long i = (long long)blockIdx.x * blockDim.x + threadIdx.x;
         i < n4; i += stride) {
        v4f m = __builtin_nontemporal_load(&mu4[i]);   // stream: TH=NT, don't pollute L2
        v4f s = __builtin_nontemporal_load(&sg4[i]);
        v4f t = __builtin_nontemporal_load(&ty4[i]);
        p += elem_f(m[0], s[0], t[0]);
        p += elem_f(m[1], s[1], t[1]);
        p += elem_f(m[2], s[2], t[2]);
        p += elem_f(m[3], s[3], t[3]);
    }

    // scalar tail (n % 4), handled by block 0 only; lanes reconverge before WMMA
    if (blockIdx.x == 0) {
        for (int i = (n & ~3) + (int)threadIdx.x; i < n; i += (int)blockDim.x) {
            p += elem_f(mu[i], sigma[i], ty[i]);
        }
    }

    // full wave active here -> EXEC all ones, WMMA legal
    float w = wave_reduce_wmma(p);

    __shared__ float sh[8];   // 256 threads = 8 wave32
    const int lane = (int)(threadIdx.x & 31u);
    const int wid  = (int)(threadIdx.x >> 5u);
    if (lane == 0) sh[wid] = w;
    __syncthreads();
    if (threadIdx.x == 0) {
        float bsum = 0.0f;
        const int nw = (int)(blockDim.x >> 5u);
        #pragma unroll
        for (int i = 0; i < 8; ++i) if (i < nw) bsum += sh[i];
        partials[blockIdx.x] = bsum;
    }
}

__global__ void gauss_nll_finalize(const float* __restrict__ partials,
                                   int nparts,
                                   float* __restrict__ out) {
    __shared__ double sh[256];
    double s = 0.0;
    for (int i = (int)threadIdx.x; i < nparts; i += (int)blockDim.x)
        s += (double)partials[i];
    sh[threadIdx.x] = s;
    __syncthreads();
    for (int off = 128; off > 0; off >>= 1) {
        if ((int)threadIdx.x < off) sh[threadIdx.x] += sh[threadIdx.x + off];
        __syncthreads();
    }
    if (threadIdx.x == 0) {
        const double LOG2PI = 1.83787706640934548356065947281;
        double total = sh[0];
        double loss = 0.5 * (double)NT_N * LOG2PI + 0.5 * total / (double)BS_N;
        out[0] = (float)loss;
    }
}

extern "C" void kernel_launch(void* const* d_in, const int* in_sizes, int n_in,
                              void* d_out, int out_size, void* d_ws, size_t ws_size,
                              hipStream_t stream) {
    const float* mu    = (const float*)d_in[0];
    const float* sigma = (const float*)d_in[1];
    const float* ty    = (const float*)d_in[2];
    float* out = (float*)d_out;
    float* partials = (float*)d_ws;   // nblocks floats of scratch

    const int n = in_sizes[0];        // BS*NT = 33,554,432

    const int threads = 256;          // 8 wave32 per block
    const int nblocks = 1024;         // 262,144 threads -> 128 elems/thread

    gauss_nll_partial<<<nblocks, threads, 0, stream>>>(mu, sigma, ty, partials, n);
    gauss_nll_finalize<<<1, 256, 0, stream>>>(partials, nblocks, out);
}